// GCNsBlock_85495618994177
// MI455X (gfx1250) — compile-verified
//
#include <hip/hip_runtime.h>
#include <hip/hip_bf16.h>
#include <cstddef>

typedef __attribute__((ext_vector_type(2))) float v2f;
typedef __attribute__((ext_vector_type(8))) float v8f;

// ---------------------------------------------------------------------------
// Fold (x@W1+b1)@W2+b2)@W3+b3 into h = x*v + u.  vu[0:64]=v, vu[64:128]=u.
__global__ __launch_bounds__(256) void k_fold_weights(
    const float* __restrict__ W1, const float* __restrict__ b1,
    const float* __restrict__ W2, const float* __restrict__ b2,
    const float* __restrict__ W3, const float* __restrict__ b3,
    float* __restrict__ vu) {
  __shared__ float vm[256], um[256];
  int j = threadIdx.x;               // 0..255 : columns of W2
  float v = 0.f, u = 0.f;
  for (int k = 0; k < 64; ++k) {
    float w2 = W2[k * 256 + j];
    v += W1[k] * w2;
    u += b1[k] * w2;
  }
  vm[j] = v; um[j] = u + b2[j];
  __syncthreads();
  if (j < 64) {
    float vv = 0.f, uu = 0.f;
    for (int t = 0; t < 256; ++t) {
      float w3 = W3[t * 64 + j];
      vv += vm[t] * w3; uu += um[t] * w3;
    }
    vu[j]      = vv;
    vu[64 + j] = uu + b3[j];
  }
}

__global__ __launch_bounds__(256) void k_deg_init(float* deg, int Nn) {
  int i = blockIdx.x * 256 + threadIdx.x;
  if (i < Nn) deg[i] = 1.0f;                      // self-loop
}

__global__ __launch_bounds__(256) void k_deg_acc(const int* __restrict__ dst,
                                                 float* deg, int E) {
  int e = blockIdx.x * 256 + threadIdx.x;
  if (e < E) atomicAdd(&deg[dst[e]], 1.0f);
}

__global__ __launch_bounds__(256) void k_dinv(float* deg, int Nn) {
  int i = blockIdx.x * 256 + threadIdx.x;
  if (i < Nn) { float d = deg[i]; deg[i] = 1.0f / sqrtf(d); }
}

// per-edge symmetric norm: w[e] = dinv[src[e]] * dinv[dst[e]]
__global__ __launch_bounds__(256) void k_ew(const int* __restrict__ src,
                                            const int* __restrict__ dst,
                                            const float* __restrict__ dinv,
                                            float* __restrict__ ew, int E) {
  int e = blockIdx.x * 256 + threadIdx.x;
  if (e < E) ew[e] = dinv[src[e]] * dinv[dst[e]];
}

// h = x*v + u ; agg1 = h * dinv^2   (self-loop contribution, direct store)
__global__ __launch_bounds__(256) void k_h_init(
    const float* __restrict__ x, const float* __restrict__ vu,
    const float* __restrict__ dinv,
    float* __restrict__ h, float* __restrict__ agg1, int Nn) {
  int idx = blockIdx.x * 256 + threadIdx.x;
  if (idx >= Nn * 64) return;
  int n = idx >> 6, c = idx & 63;
  float hv = x[n] * vu[c] + vu[64 + c];
  h[idx] = hv;
  float di = dinv[n];
  agg1[idx] = hv * di * di;
}

// conv1 scatter: agg1[d] += h[s] * w[e]   (4 channels / thread)
__global__ __launch_bounds__(256) void k_edge1(
    const int* __restrict__ src, const int* __restrict__ dst,
    const float* __restrict__ ew, const float* __restrict__ h,
    float* __restrict__ agg1, int E) {
  int idx = blockIdx.x * 256 + threadIdx.x;
  if (idx >= E * 16) return;
  int e = idx >> 4;
  int c = (idx & 15) << 2;
  int s = src[e], d = dst[e];
  float w = ew[e];
  float4 hv = *(const float4*)(h + (size_t)s * 64 + c);
  float* ap = agg1 + (size_t)d * 64 + c;
  atomicAdd(ap + 0, hv.x * w);
  atomicAdd(ap + 1, hv.y * w);
  atomicAdd(ap + 2, hv.z * w);
  atomicAdd(ap + 3, hv.w * w);
}

// ---------------------------------------------------------------------------
// fp32 WMMA GEMM:  D[Nn,128] = A[Nn,K] @ B[K,128] (+ bias)
// Block = 256 threads = 8 waves; wave w owns output cols [16w,16w+16);
// block b owns rows [16b,16b+16).  A tile staged in LDS (stride K+4 dwords:
// bank-conflict-free for the 16-lane fragment reads).  K stepped by 4 via
// V_WMMA_F32_16X16X4_F32 (exact fp32, matches reference precision).
__global__ __launch_bounds__(256) void k_wmma_gemm(
    const float* __restrict__ A, int K,
    const float* __restrict__ Bm, const float* __restrict__ bias,
    float* __restrict__ Dm, int Nn) {
  extern __shared__ float lds[];            // 16 rows x (K+4) floats
  const int S = K + 4;                      // padded LDS row stride (dwords)

  // cooperative float4 stage of the 16 x K A tile
  const int row0 = blockIdx.x * 16;
  const int nv4 = (16 * K) >> 2;            // # float4 elements
  for (int t = threadIdx.x; t < nv4; t += 256) {
    int fl = t << 2;
    int r = fl / K, c = fl % K;             // K % 4 == 0
    float4 v = *(const float4*)(A + (size_t)(row0 + r) * K + c);
    *(float4*)(lds + r * S + c) = v;
  }
  __syncthreads();

  const int lane = threadIdx.x & 31;
  const int wv   = threadIdx.x >> 5;
  const int li   = lane & 15;
  const int half = lane >> 4;
  const int col  = (wv << 4) + li;

  v8f acc = {};
  const float* arow = lds + li * S;
  for (int k0 = 0; k0 < K; k0 += 4) {
    const int ka = k0 + (half << 1);
    v2f a, b;
    a.x = arow[ka];
    a.y = arow[ka + 1];
    b.x = Bm[(size_t)ka * 128 + col];
    b.y = Bm[(size_t)(ka + 1) * 128 + col];
    acc = __builtin_amdgcn_wmma_f32_16x16x4_f32(
        /*neg_a=*/false, a, /*neg_b=*/false, b,
        /*c_mod=*/(short)0, acc, /*reuse_a=*/false, /*reuse_b=*/false);
  }
  float bv = bias ? bias[col] : 0.0f;
#pragma unroll
  for (int v = 0; v < 8; ++v) {
    int row = row0 + (half << 3) + v;
    Dm[(size_t)row * 128 + col] = acc[v] + bv;
  }
}

// scalar tail for rows not covered by full 16-row tiles (no-op for N=40000)
__global__ __launch_bounds__(256) void k_gemm_tail(
    const float* __restrict__ A, int K, const float* __restrict__ Bm,
    const float* __restrict__ bias, float* __restrict__ Dm,
    int rowStart, int Nn) {
  int idx = blockIdx.x * 256 + threadIdx.x;
  int row = rowStart + (idx >> 7);
  int c = idx & 127;
  if (row >= Nn) return;
  float s = bias ? bias[c] : 0.0f;
  for (int k = 0; k < K; ++k) s += A[(size_t)row * K + k] * Bm[(size_t)k * 128 + c];
  Dm[(size_t)row * 128 + c] = s;
}

// z[n] = relu([h1[n], h[root[batch[n]]]])   -> [N,192]
__global__ __launch_bounds__(192) void k_zbuild(
    const float* __restrict__ h1, const float* __restrict__ h,
    const int* __restrict__ batch, const int* __restrict__ root,
    float* __restrict__ z, int Nn) {
  int n = blockIdx.x;
  int c = threadIdx.x;
  float v;
  if (c < 128) {
    v = h1[(size_t)n * 128 + c];
  } else {
    int r = root[batch[n]];
    v = h[(size_t)r * 64 + (c - 128)];
  }
  z[(size_t)n * 192 + c] = fmaxf(v, 0.0f);
}

// h2 = zW*dinv^2 + bc2   (self-loop + bias, direct store)
__global__ __launch_bounds__(256) void k_h2_init(
    const float* __restrict__ zW, const float* __restrict__ dinv,
    const float* __restrict__ bc2, float* __restrict__ h2, int Nn) {
  int idx = blockIdx.x * 256 + threadIdx.x;
  if (idx >= Nn * 128) return;
  int n = idx >> 7, c = idx & 127;
  float di = dinv[n];
  h2[idx] = zW[idx] * di * di + bc2[c];
}

// conv2 scatter: h2[d] += zW[s] * w[e]
__global__ __launch_bounds__(256) void k_edge2(
    const int* __restrict__ src, const int* __restrict__ dst,
    const float* __restrict__ ew, const float* __restrict__ zW,
    float* __restrict__ h2, int E) {
  int idx = blockIdx.x * 256 + threadIdx.x;
  if (idx >= E * 32) return;
  int e = idx >> 5;
  int c = (idx & 31) << 2;
  int s = src[e], d = dst[e];
  float w = ew[e];
  float4 v = *(const float4*)(zW + (size_t)s * 128 + c);
  float* p = h2 + (size_t)d * 128 + c;
  atomicAdd(p + 0, v.x * w);
  atomicAdd(p + 1, v.y * w);
  atomicAdd(p + 2, v.z * w);
  atomicAdd(p + 3, v.w * w);
}

__global__ __launch_bounds__(256) void k_cnt(const int* __restrict__ batch,
                                             float* cnt, int Nn) {
  int i = blockIdx.x * 256 + threadIdx.x;
  if (i < Nn) atomicAdd(&cnt[batch[i]], 1.0f);
}

__global__ __launch_bounds__(256) void k_acc(const float* __restrict__ h2,
                                             const int* __restrict__ batch,
                                             float* acc, int Nn) {
  int idx = blockIdx.x * 256 + threadIdx.x;
  if (idx >= Nn * 128) return;
  int n = idx >> 7, c = idx & 127;
  atomicAdd(&acc[(size_t)batch[n] * 128 + c], fmaxf(h2[idx], 0.0f));
}

// out[b,0:128]=sum relu(h2)/max(cnt,1) ; out[b,128:256]=h1[root[b]] (cnt>0)
__global__ __launch_bounds__(256) void k_final(
    const float* __restrict__ acc, const float* __restrict__ cnt,
    const float* __restrict__ h1, const int* __restrict__ root,
    float* __restrict__ out, int B) {
  int b = blockIdx.x, c = threadIdx.x;
  float cb = cnt[b];
  float v;
  if (c < 128) v = acc[(size_t)b * 128 + c] / fmaxf(cb, 1.0f);
  else         v = (cb > 0.0f) ? h1[(size_t)root[b] * 128 + (c - 128)] : 0.0f;
  out[(size_t)b * 256 + c] = v;
}

// ---------------------------------------------------------------------------
extern "C" void kernel_launch(void* const* d_in, const int* in_sizes, int n_in,
                              void* d_out, int out_size, void* d_ws, size_t ws_size,
                              hipStream_t stream) {
  const float* x     = (const float*)d_in[0];
  const int*   ei    = (const int*)d_in[1];
  const int*   batch = (const int*)d_in[2];
  const int*   root  = (const int*)d_in[3];
  const float* W1 = (const float*)d_in[4];
  const float* b1 = (const float*)d_in[5];
  const float* W2 = (const float*)d_in[6];
  const float* b2 = (const float*)d_in[7];
  const float* W3 = (const float*)d_in[8];
  const float* b3 = (const float*)d_in[9];
  const float* Wc1 = (const float*)d_in[10];
  const float* bc1 = (const float*)d_in[11];
  const float* Wc2 = (const float*)d_in[12];
  const float* bc2 = (const float*)d_in[13];

  const int Nn = in_sizes[0];       // x is [N,1]
  const int E  = in_sizes[1] / 2;   // edge_index [2,E]
  const int B  = in_sizes[3];       // root_index [B]
  const int* src = ei;
  const int* dst = ei + E;

  float* ws = (float*)d_ws;
  size_t o = 0;
  auto bump = [&](size_t nf) { float* p = ws + o; o = (o + nf + 3) & ~(size_t)3; return p; };
  float* dinv = bump((size_t)Nn);          // deg -> dinv in place
  float* ewt  = bump((size_t)E);           // per-edge norm
  float* h    = bump((size_t)Nn * 64);
  float* agg1 = bump((size_t)Nn * 64);
  float* h1   = bump((size_t)Nn * 128);
  float* z    = bump((size_t)Nn * 192);
  float* zW   = bump((size_t)Nn * 128);
  float* h2   = bump((size_t)Nn * 128);
  float* vu   = bump(128);
  float* acc  = bump((size_t)B * 128);
  float* cnt  = bump((size_t)B);

  auto cdiv = [](long long a, long long b) { return (int)((a + b - 1) / b); };

  hipMemsetAsync(acc, 0, sizeof(float) * (size_t)B * 128, stream);
  hipMemsetAsync(cnt, 0, sizeof(float) * (size_t)B, stream);

  k_fold_weights<<<1, 256, 0, stream>>>(W1, b1, W2, b2, W3, b3, vu);
  k_deg_init<<<cdiv(Nn, 256), 256, 0, stream>>>(dinv, Nn);
  k_deg_acc<<<cdiv(E, 256), 256, 0, stream>>>(dst, dinv, E);
  k_dinv<<<cdiv(Nn, 256), 256, 0, stream>>>(dinv, Nn);
  k_ew<<<cdiv(E, 256), 256, 0, stream>>>(src, dst, dinv, ewt, E);
  k_h_init<<<cdiv((long long)Nn * 64, 256), 256, 0, stream>>>(x, vu, dinv, h, agg1, Nn);
  k_edge1<<<cdiv((long long)E * 16, 256), 256, 0, stream>>>(src, dst, ewt, h, agg1, E);

  int tiles = Nn / 16;
  int rem = Nn - tiles * 16;
  size_t sh1 = sizeof(float) * 16 * (64 + 4);
  size_t sh2 = sizeof(float) * 16 * (192 + 4);
  if (tiles > 0)
    k_wmma_gemm<<<tiles, 256, sh1, stream>>>(agg1, 64, Wc1, bc1, h1, Nn);
  if (rem > 0)
    k_gemm_tail<<<cdiv((long long)rem * 128, 256), 256, 0, stream>>>(agg1, 64, Wc1, bc1, h1, tiles * 16, Nn);

  k_zbuild<<<Nn, 192, 0, stream>>>(h1, h, batch, root, z, Nn);

  if (tiles > 0)
    k_wmma_gemm<<<tiles, 256, sh2, stream>>>(z, 192, Wc2, nullptr, zW, Nn);
  if (rem > 0)
    k_gemm_tail<<<cdiv((long long)rem * 128, 256), 256, 0, stream>>>(z, 192, Wc2, nullptr, zW, tiles * 16, Nn);

  k_h2_init<<<cdiv((long long)Nn * 128, 256), 256, 0, stream>>>(zW, dinv, bc2, h2, Nn);
  k_edge2<<<cdiv((long long)E * 32, 256), 256, 0, stream>>>(src, dst, ewt, zW, h2, E);
  k_cnt<<<cdiv(Nn, 256), 256, 0, stream>>>(batch, cnt, Nn);
  k_acc<<<cdiv((long long)Nn * 128, 256), 256, 0, stream>>>(h2, batch, acc, Nn);
  k_final<<<B, 256, 0, stream>>>(acc, cnt, h1, root, (float*)d_out, B);
}